// DetectionLoss_89567247991290
// MI455X (gfx1250) — compile-verified
//
#include <hip/hip_runtime.h>

// ---------------- problem constants (from reference) ----------------
static constexpr int B = 16, A = 9, C = 21, H = 80, W = 80, T = 20;
static constexpr int HW = H * W;            // 6400
static constexpr int N = A * HW;            // 57600 anchors per image
static constexpr int CHUNK = 256;           // anchors per block (6400 % 256 == 0)
static constexpr int NCHUNK = N / CHUNK;    // 225 blocks per image
static constexpr float BOX_REG_SCALE = 0.1f;
static constexpr float POS_IOU = 0.5f;
static constexpr float NEG_IOU = 0.4f;
static constexpr float IMG_SIZE = 512.0f;
static constexpr int MIN_POS_TOPK = 10;

// ---------------- CDNA5 TDM (tensor_load_to_lds) --------------------
#if defined(__gfx1250__) && __has_builtin(__builtin_amdgcn_tensor_load_to_lds) && \
    __has_builtin(__builtin_amdgcn_s_wait_tensorcnt)
#define USE_TDM 1
#else
#define USE_TDM 0
#endif

#if USE_TDM
typedef unsigned int u32x4 __attribute__((ext_vector_type(4)));
typedef int i32x4 __attribute__((ext_vector_type(4)));
typedef int i32x8 __attribute__((ext_vector_type(8)));

// 2D tile load: `rows` rows of `row_elems` f32, row stride `stride_elems` f32,
// from global `gptr` into LDS byte offset `lds_byte`. Built per D# spec
// (CDNA5 ISA). tensor_dim == tile_dim -> no OOB.
__device__ __forceinline__ void tdm_load_2d(unsigned lds_byte, const void* gptr,
                                            unsigned rows, unsigned row_elems,
                                            unsigned stride_elems) {
  unsigned long long ga = (unsigned long long)gptr;
  u32x4 g0;
  g0[0] = 1u;                                            // count=1, user mode
  g0[1] = lds_byte;                                      // lds_addr (bytes)
  g0[2] = (unsigned)(ga & 0xFFFFFFFFu);                  // global_addr[31:0]
  g0[3] = ((unsigned)(ga >> 32) & 0x01FFFFFFu) | 0x80000000u; // addr[56:32], type=2
  i32x8 g1;
  g1[0] = (int)(2u << 16);                               // data_size = 4B
  g1[1] = (int)((row_elems & 0xFFFFu) << 16);            // tensor_dim0[15:0]
  g1[2] = (int)(((row_elems >> 16) & 0xFFFFu) | ((rows & 0xFFFFu) << 16)); // dim0 hi | tensor_dim1 lo
  g1[3] = (int)(((rows >> 16) & 0xFFFFu) | ((row_elems & 0xFFFFu) << 16)); // dim1 hi | tile_dim0
  g1[4] = (int)(rows & 0xFFFFu);                         // tile_dim1 (tile_dim2=0)
  g1[5] = (int)stride_elems;                             // tensor_dim0_stride[31:0]
  g1[6] = 0;
  g1[7] = 0;
  i32x4 z4 = {0, 0, 0, 0};
#if __has_include(<hip/amd_detail/amd_gfx1250_TDM.h>)
  i32x8 z8 = {0, 0, 0, 0, 0, 0, 0, 0};
  __builtin_amdgcn_tensor_load_to_lds(g0, g1, z4, z4, z8, 0); // 6-arg toolchain
#else
  __builtin_amdgcn_tensor_load_to_lds(g0, g1, z4, z4, 0);     // ROCm 7.2 5-arg
#endif
}

__device__ __forceinline__ unsigned lds_off(const void* p) {
  // generic pointer to LDS: low 32 bits are the LDS byte offset
  return (unsigned)(unsigned long long)p;
}
#endif  // USE_TDM

// ---------------- helpers -------------------------------------------
__device__ __forceinline__ float warp_sum(float v) {
#pragma unroll
  for (int off = 16; off; off >>= 1) v += __shfl_down(v, off, 32);
  return v;
}

__device__ __forceinline__ void decode_box(float4 a, float r0, float r1, float r2,
                                           float r3, float d[4]) {
  float aw = a.z - a.x, ah = a.w - a.y;
  float acx = a.x + 0.5f * aw, acy = a.y + 0.5f * ah;
  float d0 = r0 * BOX_REG_SCALE, d1 = r1 * BOX_REG_SCALE;
  float d2 = fminf(fmaxf(r2 * BOX_REG_SCALE, -4.0f), 4.0f);
  float d3 = fminf(fmaxf(r3 * BOX_REG_SCALE, -4.0f), 4.0f);
  float cx = acx + d0 * aw, cy = acy + d1 * ah;
  float w = aw * expf(d2), h = ah * expf(d3);
  d[0] = cx - 0.5f * w; d[1] = cy - 0.5f * h;
  d[2] = cx + 0.5f * w; d[3] = cy + 0.5f * h;
}

// ---------------- K1: decode + pairwise IoU max/argmax --------------
__global__ __launch_bounds__(256) void k_iou(const float* __restrict__ reg,
                                             const float* __restrict__ anchors,
                                             const float* __restrict__ tboxes,
                                             float* __restrict__ maxiou,
                                             int* __restrict__ maxidx) {
  __shared__ float reg_s[4 * CHUNK];
  __shared__ float tb_s[T * 4];
  const int b = blockIdx.y, chunk = blockIdx.x, tid = threadIdx.x;
  const int n0 = chunk * CHUNK;
  const int a = n0 / HW, hw0 = n0 % HW;
  const float* greg = reg + ((size_t)(b * A + a) * 4) * HW + hw0;

  if (tid >= 64 && tid < 64 + T * 4) tb_s[tid - 64] = tboxes[(size_t)b * T * 4 + (tid - 64)];
#if USE_TDM
  if (tid < 32) {
    tdm_load_2d(lds_off(reg_s), greg, 4, CHUNK, HW);
    __builtin_amdgcn_s_wait_tensorcnt(0);
  }
#else
  for (int i = tid; i < 4 * CHUNK; i += 256)
    reg_s[i] = greg[(size_t)(i >> 8) * HW + (i & (CHUNK - 1))];
#endif
  __syncthreads();

  const int n = n0 + tid;
  float4 a4 = reinterpret_cast<const float4*>(anchors)[n];
  float d[4];
  decode_box(a4, reg_s[tid], reg_s[CHUNK + tid], reg_s[2 * CHUNK + tid],
             reg_s[3 * CHUNK + tid], d);
  float areaA = (d[2] - d[0]) * (d[3] - d[1]);
  float best = -1.0f; int bidx = 0;
#pragma unroll
  for (int t = 0; t < T; ++t) {
    float tx1 = tb_s[4 * t + 0], ty1 = tb_s[4 * t + 1];
    float tx2 = tb_s[4 * t + 2], ty2 = tb_s[4 * t + 3];
    float iw = fmaxf(fminf(d[2], tx2) - fmaxf(d[0], tx1), 0.0f);
    float ih = fmaxf(fminf(d[3], ty2) - fmaxf(d[1], ty1), 0.0f);
    float inter = iw * ih;
    float areaB = (tx2 - tx1) * (ty2 - ty1);
    float iou = inter / (areaA + areaB - inter + 1e-8f);
    if (iou > best) { best = iou; bidx = t; }   // first-max tie-break (jnp.argmax)
  }
  maxiou[(size_t)b * N + n] = best;
  maxidx[(size_t)b * N + n] = bidx;
}

// ---------------- K2: per-image pos count + exact top-10 ------------
__global__ __launch_bounds__(1024) void k_stats(const float* __restrict__ maxiou,
                                                int* __restrict__ stats) {
  const int b = blockIdx.x, tid = threadIdx.x;
  const float* mi = maxiou + (size_t)b * N;
  int* st = stats + b * 12;  // [0]=pos_count [1]=use_topk [2..11]=topk idx
  __shared__ int s_cnt[32];
  __shared__ float s_v[32]; __shared__ int s_n[32];
  __shared__ float s_pv; __shared__ int s_pn; __shared__ int s_use;

  int cnt = 0;
  for (int n = tid; n < N; n += 1024) cnt += (mi[n] >= POS_IOU) ? 1 : 0;
#pragma unroll
  for (int off = 16; off; off >>= 1) cnt += __shfl_down(cnt, off, 32);
  if ((tid & 31) == 0) s_cnt[tid >> 5] = cnt;
  __syncthreads();
  if (tid < 32) {
    int c = s_cnt[tid];
#pragma unroll
    for (int off = 16; off; off >>= 1) c += __shfl_down(c, off, 32);
    if (tid == 0) { st[0] = c; s_use = (c < MIN_POS_TOPK) ? 1 : 0; st[1] = s_use; }
  }
  __syncthreads();

  for (int k = 0; k < MIN_POS_TOPK; ++k) {
    float bv = -INFINITY; int bn = N;
    if (s_use) {
      float pv = s_pv; int pn = s_pn;
      for (int n = tid; n < N; n += 1024) {
        float v = mi[n];
        bool adm = (k == 0) || (v < pv) || (v == pv && n > pn);
        if (adm && (v > bv || (v == bv && n < bn))) { bv = v; bn = n; }
      }
    }
#pragma unroll
    for (int off = 16; off; off >>= 1) {
      float ov = __shfl_down(bv, off, 32); int on = __shfl_down(bn, off, 32);
      if (ov > bv || (ov == bv && on < bn)) { bv = ov; bn = on; }
    }
    if ((tid & 31) == 0) { s_v[tid >> 5] = bv; s_n[tid >> 5] = bn; }
    __syncthreads();
    if (tid < 32) {
      bv = s_v[tid]; bn = s_n[tid];
#pragma unroll
      for (int off = 16; off; off >>= 1) {
        float ov = __shfl_down(bv, off, 32); int on = __shfl_down(bn, off, 32);
        if (ov > bv || (ov == bv && on < bn)) { bv = ov; bn = on; }
      }
      if (tid == 0) { st[2 + k] = s_use ? bn : -1; s_pv = bv; s_pn = bn; }
    }
    __syncthreads();
  }
}

// ---------------- K3: main loss pass (TDM-fed) ----------------------
__global__ __launch_bounds__(256) void k_main(
    const float* __restrict__ cls, const float* __restrict__ reg,
    const float* __restrict__ anchors, const float* __restrict__ tboxes,
    const int* __restrict__ tlabels, const float* __restrict__ maxiou,
    const int* __restrict__ maxidx, const int* __restrict__ stats,
    float* __restrict__ partial) {
  __shared__ float cls_s[C * CHUNK];  // 21.0 KB
  __shared__ float reg_s[4 * CHUNK];  //  4.0 KB
  __shared__ float tb_s[T * 4];
  __shared__ int tl_s[T];
  __shared__ int tk_s[MIN_POS_TOPK];
  __shared__ int s_use;
  __shared__ float red[8][5];

  const int b = blockIdx.y, chunk = blockIdx.x, tid = threadIdx.x;
  const int n0 = chunk * CHUNK;
  const int a = n0 / HW, hw0 = n0 % HW;
  const float* gcls = cls + ((size_t)(b * A + a) * C) * HW + hw0;
  const float* greg = reg + ((size_t)(b * A + a) * 4) * HW + hw0;

  if (tid >= 64 && tid < 64 + T * 4) tb_s[tid - 64] = tboxes[(size_t)b * T * 4 + (tid - 64)];
  else if (tid >= 160 && tid < 160 + T) tl_s[tid - 160] = tlabels[b * T + (tid - 160)];
  else if (tid >= 192 && tid < 192 + MIN_POS_TOPK) tk_s[tid - 192] = stats[b * 12 + 2 + (tid - 192)];
  else if (tid == 224) s_use = stats[b * 12 + 1];
#if USE_TDM
  if (tid < 32) {  // one wave issues both async DMA descriptors
    tdm_load_2d(lds_off(cls_s), gcls, C, CHUNK, HW);
    tdm_load_2d(lds_off(reg_s), greg, 4, CHUNK, HW);
    __builtin_amdgcn_s_wait_tensorcnt(0);
  }
#else
  for (int i = tid; i < C * CHUNK; i += 256)
    cls_s[i] = gcls[(size_t)(i >> 8) * HW + (i & (CHUNK - 1))];
  for (int i = tid; i < 4 * CHUNK; i += 256)
    reg_s[i] = greg[(size_t)(i >> 8) * HW + (i & (CHUNK - 1))];
#endif
  __syncthreads();

  const int n = n0 + tid;
  // log-softmax over C=21 (max-shifted), logits from LDS (conflict-free)
  float m = -INFINITY;
#pragma unroll
  for (int c = 0; c < C; ++c) m = fmaxf(m, cls_s[c * CHUNK + tid]);
  float s = 0.0f;
#pragma unroll
  for (int c = 0; c < C; ++c) s += expf(cls_s[c * CHUNK + tid] - m);
  float logZ = m + logf(s);

  float iou = maxiou[(size_t)b * N + n];
  int midx = maxidx[(size_t)b * N + n];
  bool pos;
  if (s_use) {
    pos = false;
#pragma unroll
    for (int k = 0; k < MIN_POS_TOPK; ++k) pos = pos || (tk_s[k] == n);
  } else {
    pos = (iou >= POS_IOU);
  }
  bool neg = (iou < NEG_IOU);

  float ce_pos = logZ - cls_s[tl_s[midx] * CHUNK + tid];  // dynamic class via LDS
  float ce_neg = logZ - cls_s[tid];                       // background class 0

  float4 a4 = reinterpret_cast<const float4*>(anchors)[n];
  float d[4];
  decode_box(a4, reg_s[tid], reg_s[CHUNK + tid], reg_s[2 * CHUNK + tid],
             reg_s[3 * CHUNK + tid], d);
  float l1 = 0.0f;
#pragma unroll
  for (int q = 0; q < 4; ++q) l1 += fabsf(d[q] - tb_s[midx * 4 + q]);
  l1 *= (0.25f / IMG_SIZE);

  float posf = pos ? 1.0f : 0.0f, negf = neg ? 1.0f : 0.0f;
  float vals[5] = {ce_pos * posf, ce_neg * negf, l1 * posf, posf, negf};
#pragma unroll
  for (int q = 0; q < 5; ++q) vals[q] = warp_sum(vals[q]);
  if ((tid & 31) == 0) {
#pragma unroll
    for (int q = 0; q < 5; ++q) red[tid >> 5][q] = vals[q];
  }
  __syncthreads();
  if (tid == 0) {
    float acc[5] = {0, 0, 0, 0, 0};
    for (int w2 = 0; w2 < 8; ++w2)
      for (int q = 0; q < 5; ++q) acc[q] += red[w2][q];
    float* p = partial + ((size_t)b * NCHUNK + chunk) * 5;
    for (int q = 0; q < 5; ++q) p[q] = acc[q];
  }
}

// ---------------- K4: deterministic per-image reduce ----------------
__global__ __launch_bounds__(256) void k_reduce(const float* __restrict__ partial,
                                                float* __restrict__ accum) {
  __shared__ float red[8][5];
  const int b = blockIdx.x, tid = threadIdx.x;
  float v[5] = {0, 0, 0, 0, 0};
  if (tid < NCHUNK) {
    const float* p = partial + ((size_t)b * NCHUNK + tid) * 5;
#pragma unroll
    for (int q = 0; q < 5; ++q) v[q] = p[q];
  }
#pragma unroll
  for (int q = 0; q < 5; ++q) v[q] = warp_sum(v[q]);
  if ((tid & 31) == 0) {
#pragma unroll
    for (int q = 0; q < 5; ++q) red[tid >> 5][q] = v[q];
  }
  __syncthreads();
  if (tid == 0) {
    float acc[5] = {0, 0, 0, 0, 0};
    for (int w2 = 0; w2 < 8; ++w2)
      for (int q = 0; q < 5; ++q) acc[q] += red[w2][q];
    for (int q = 0; q < 5; ++q) accum[b * 5 + q] = acc[q];
  }
}

// ---------------- K5: final scalars ---------------------------------
__global__ void k_final(const float* __restrict__ accum, float* __restrict__ out) {
  if (threadIdx.x == 0 && blockIdx.x == 0) {
    float cs = 0.0f, rs = 0.0f, tp = 0.0f;
    for (int b = 0; b < B; ++b) {
      float sp = accum[b * 5 + 0], sn = accum[b * 5 + 1], sl = accum[b * 5 + 2];
      float cp = accum[b * 5 + 3], cn = accum[b * 5 + 4];
      float cl = ((cp > 0.0f) ? sp / cp : 0.0f) + ((cn > 0.0f) ? sn / cn : 0.0f);
      float rl = (cp > 0.0f) ? sl / cp : 0.0f;
      cs += cl; rs += rl; tp += cp;
    }
    float cf = cs / (float)B;
    float rf = (tp > 0.0f) ? rs / (float)B : 0.0f;
    out[0] = cf + rf;   // total
    out[1] = cf;        // cls_final
    out[2] = rf;        // reg_final
    out[3] = tp;        // total_pos
  }
}

// ---------------- launch --------------------------------------------
extern "C" void kernel_launch(void* const* d_in, const int* in_sizes, int n_in,
                              void* d_out, int out_size, void* d_ws, size_t ws_size,
                              hipStream_t stream) {
  (void)in_sizes; (void)n_in; (void)out_size; (void)ws_size;
  const float* cls = (const float*)d_in[0];
  const float* reg = (const float*)d_in[1];
  const float* anch = (const float*)d_in[2];
  const float* tbox = (const float*)d_in[3];
  const int* tlab = (const int*)d_in[4];
  float* out = (float*)d_out;

  char* ws = (char*)d_ws;
  const size_t BN = (size_t)B * N;
  float* maxiou = (float*)ws;                              // BN f32
  int* maxidx = (int*)(ws + BN * 4);                       // BN i32
  int* stats = (int*)(ws + 2 * BN * 4);                    // B*12 i32
  float* partial = (float*)(ws + 2 * BN * 4 + B * 12 * 4); // B*NCHUNK*5 f32
  float* accum = (float*)(ws + 2 * BN * 4 + B * 12 * 4 +
                          (size_t)B * NCHUNK * 5 * 4);     // B*5 f32

  dim3 grid(NCHUNK, B);
  k_iou<<<grid, 256, 0, stream>>>(reg, anch, tbox, maxiou, maxidx);
  k_stats<<<B, 1024, 0, stream>>>(maxiou, stats);
  k_main<<<grid, 256, 0, stream>>>(cls, reg, anch, tbox, tlab, maxiou, maxidx,
                                   stats, partial);
  k_reduce<<<B, 256, 0, stream>>>(partial, accum);
  k_final<<<1, 32, 0, stream>>>(accum, out);
}